// _EquiAttention_11931419148535
// MI455X (gfx1250) — compile-verified
//
#include <hip/hip_runtime.h>
#include <hip/hip_bf16.h>
#include <math.h>

// Sizes fixed by the reference: B=16, NQ=256, NKV=256, D=64.
#define BB   16
#define NQ   256
#define NKV  256
#define DD   64
#define MROWS (BB * NKV * 3)   // 12288 rows for both GEMMs

typedef float v2f __attribute__((ext_vector_type(2)));
typedef float v8f __attribute__((ext_vector_type(8)));

// ---------------------------------------------------------------------------
// Kernel A: C[M,64] = A[M,64] @ W[64,64]^T  using V_WMMA_F32_16X16X4_F32.
// One wave (32 lanes) per 16x16 C tile; 16 chained WMMAs over K=64.
//
// ISA 7.12.2 f32 layouts (wave32):
//   A 16x4 (MxK): lanes 0-15 row m, VGPR0=K0,VGPR1=K1; lanes 16-31: K2,K3.
//   B 4x16 (KxN): lane (&15) = column n; lanes 0-15 hold K0/K1, 16-31 K2/K3.
//   C 16x16: VGPR r -> row (r + 8*(lane>=16)), col = lane&15.
// B[k][n] = W[n][k]  (because C = A @ W^T), so each lane's (b.x,b.y) is a
// contiguous 8B pair from W's row n -> single global_load_b64 per fragment.
// ---------------------------------------------------------------------------
__global__ __launch_bounds__(32)
void gemm_f32_wmma_kernel(const float* __restrict__ A,
                          const float* __restrict__ W,
                          float* __restrict__ C) {
    const int lane = threadIdx.x;        // 0..31
    const int half = lane >> 4;          // 0 or 1
    const int l16  = lane & 15;
    const int m0 = blockIdx.x * 16;
    const int n0 = blockIdx.y * 16;

    const float* __restrict__ Arow = A + (size_t)(m0 + l16) * DD;
    const float* __restrict__ Wrow = W + (size_t)(n0 + l16) * DD;

    v8f c = {};
#pragma unroll
    for (int k0 = 0; k0 < DD; k0 += 4) {
        const int kb = k0 + half * 2;
        v2f a = *(const v2f*)(Arow + kb);   // A[m][kb], A[m][kb+1]
        v2f b = *(const v2f*)(Wrow + kb);   // W[n][kb], W[n][kb+1] = B[kb..][n]
        // 8 args: (neg_a, A, neg_b, B, c_mod, C, reuse_a, reuse_b)
        c = __builtin_amdgcn_wmma_f32_16x16x4_f32(
                false, a, false, b, (short)0, c, false, false);
    }

    float* __restrict__ Crow = C + (size_t)(m0 + half * 8) * DD + n0 + l16;
#pragma unroll
    for (int r = 0; r < 8; ++r)
        Crow[(size_t)r * DD] = c[r];
}

// ---------------------------------------------------------------------------
// Kernel B: fused masked online-softmax over k, Σk attn*proj (3 channels),
// and Σk attn^2 (row L2 weight). One block per (b,q); thread t = channel d.
// Reads the 268MB `messages` tensor exactly once, fully coalesced.
// Writes scaled[b,q,c,d] = acc_c * sqrt(ss) / s^2  (== attn_out * weights).
// ---------------------------------------------------------------------------
__global__ __launch_bounds__(64)
void equi_attn_fused_kernel(const float* __restrict__ messages,
                            const int*   __restrict__ adj,
                            const float* __restrict__ proj,
                            float*       __restrict__ scaled) {
    __shared__ float maskf[NKV];
    __shared__ int   s_any;

    const int t  = threadIdx.x;          // d channel, 0..63
    const int bq = blockIdx.x;           // 0..4095
    const int b  = bq >> 8;

    if (t == 0) s_any = 0;
    __syncthreads();

    // Build additive mask (-1e30 avoids -inf - -inf = NaN in online rescale),
    // and detect empty rows (reference maps empty rows to mask==0).
    int myany = 0;
#pragma unroll
    for (int i = 0; i < 4; ++i) {
        const int k = t + i * 64;
        const int a = adj[(size_t)bq * NKV + k];
        maskf[k] = a ? 0.0f : -1e30f;
        myany |= a;
    }
    if (myany) atomicOr(&s_any, 1);
    __syncthreads();
    if (!s_any) {
#pragma unroll
        for (int i = 0; i < 4; ++i) maskf[t + i * 64] = 0.0f;
    }
    __syncthreads();

    const float* __restrict__ msgp  = messages + (size_t)bq * NKV * DD + t;
    const float* __restrict__ projp = proj + (size_t)b * NKV * 3 * DD + t;

    // Online softmax state (branchless rescale; x - mn <= 0 so no overflow).
    float m = -INFINITY;
    float s = 0.0f, ss = 0.0f;
    float a0 = 0.0f, a1 = 0.0f, a2 = 0.0f;

#pragma unroll 4
    for (int k = 0; k < NKV; ++k) {
        const float x    = msgp[(size_t)k * DD] + maskf[k];
        const float mn   = fmaxf(m, x);
        const float cold = __expf(m - mn);   // 0 on first iter (exp(-inf))
        const float p    = __expf(x - mn);
        m = mn;
        const float* __restrict__ pp = projp + (size_t)k * (3 * DD);
        const float p0 = pp[0], p1 = pp[DD], p2 = pp[2 * DD];
        const float c2 = cold * cold;
        s  = s  * cold + p;
        ss = ss * c2   + p * p;
        a0 = a0 * cold + p * p0;
        a1 = a1 * cold + p * p1;
        a2 = a2 * cold + p * p2;
    }

    // attn_out * weights = (acc/s) * (sqrt(ss)/s) = acc * sqrt(ss) / s^2
    const float scale = __fsqrt_rn(ss) / (s * s);
    float* __restrict__ outp = scaled + (size_t)bq * (3 * DD) + t;
    outp[0]      = a0 * scale;
    outp[DD]     = a1 * scale;
    outp[2 * DD] = a2 * scale;
}

// ---------------------------------------------------------------------------
extern "C" void kernel_launch(void* const* d_in, const int* in_sizes, int n_in,
                              void* d_out, int out_size, void* d_ws, size_t ws_size,
                              hipStream_t stream) {
    const float* v_equi   = (const float*)d_in[0];  // [B,NKV,3,D]
    const float* messages = (const float*)d_in[1];  // [B,NQ,NKV,D]
    const int*   adj      = (const int*)  d_in[2];  // [B,NQ,NKV]
    const float* W_coord  = (const float*)d_in[3];  // [D,D]
    const float* W_attn   = (const float*)d_in[4];  // [D,D]
    float*       out      = (float*)d_out;          // [B,NQ,3,D]

    float* proj   = (float*)d_ws;                   // [B,NKV,3,D]  (3MB)
    float* scaled = proj + (size_t)MROWS * DD;      // [B,NQ,3,D]   (3MB)

    dim3 ggrid(MROWS / 16, DD / 16);                // 768 x 4 tiles

    // 1) proj_equi = v_equi @ W_coord^T   (WMMA f32)
    gemm_f32_wmma_kernel<<<ggrid, 32, 0, stream>>>(v_equi, W_coord, proj);

    // 2) fused masked online softmax + weighted accumulation + L2 weight
    equi_attn_fused_kernel<<<dim3(BB * NQ), 64, 0, stream>>>(
        messages, adj, proj, scaled);

    // 3) out = scaled @ W_attn^T          (WMMA f32)
    gemm_f32_wmma_kernel<<<ggrid, 32, 0, stream>>>(scaled, W_attn, out);
}